// FixedPointLayer_72782515799018
// MI455X (gfx1250) — compile-verified
//
#include <hip/hip_runtime.h>

typedef __attribute__((ext_vector_type(2))) float v2f;
typedef __attribute__((ext_vector_type(8))) float v8f;

#define N_DIM     64
#define BM        128          // batch rows per workgroup (8 waves x 16 rows)
#define KRICH     48           // Richardson sweeps for the Newton solve
#define SWSTRIDE  66           // padded LDS row stride (dwords) -> no bank conflicts

// gfx1250 has a hardware V_TANH_F32 transcendental (TRANS pipe, co-executes
// with WMMA). Use it if the toolchain exposes the builtin; else ocml tanhf.
__device__ __forceinline__ float fast_tanh(float v) {
#if __has_builtin(__builtin_amdgcn_tanhf)
  return __builtin_amdgcn_tanhf(v);
#elif __has_builtin(__builtin_amdgcn_tanh_f32)
  return __builtin_amdgcn_tanh_f32(v);
#else
  return tanhf(v);
#endif
}

// One wave computes Y(16x64) += A(16x64) @ B(64x64) using V_WMMA_F32_16X16X4_F32.
// A: wave-private LDS, row-major 16 x SWSTRIDE.
// Bf: 2048 float2 B-fragments, indexed [(nt*16+kc)*32 + lane] (built once from W).
__device__ __forceinline__ void gemm_16x64(const float* __restrict__ A,
                                           const float2* __restrict__ Bf,
                                           int lane, v8f* __restrict__ acc) {
  const int hi = lane >> 4;   // half-wave selector
  const int lo = lane & 15;   // row (A) / col (B,C,D) within tile
#pragma unroll
  for (int kc = 0; kc < 16; ++kc) {
    const int k0 = 4 * kc + 2 * hi;
    // A fragment (ISA 32-bit A 16x4 layout): lane holds row=lo, K = k0, k0+1
    float2 av = *(const float2*)(A + lo * SWSTRIDE + k0);
    v2f a; a.x = av.x; a.y = av.y;
#pragma unroll
    for (int nt = 0; nt < 4; ++nt) {
      // B fragment (4x16): lane holds col=lo of tile nt, K = k0, k0+1
      float2 bv = Bf[(nt * 16 + kc) * 32 + lane];
      v2f b; b.x = bv.x; b.y = bv.y;
      acc[nt] = __builtin_amdgcn_wmma_f32_16x16x4_f32(
          /*neg_a=*/false, a, /*neg_b=*/false, b,
          /*c_mod=*/(short)0, acc[nt], /*reuse_a=*/false, /*reuse_b=*/false);
    }
  }
}

__global__ __launch_bounds__(256)
void FixedPointLayer_newton_wmma(const float* __restrict__ x,
                                 const float* __restrict__ W,
                                 float* __restrict__ out, int rows) {
  __shared__ float2 sB[2048];                 // 16 KB: W pre-swizzled as B-fragments
  __shared__ float  sSwap[8 * 16 * SWSTRIDE]; // 33 KB: per-wave A-layout staging

  // Build B fragments: B[k][n] = W[n][k]  (Y = Z @ W^T).
  // slot s -> (nt, kc, lane); value = (W[n*64+k], W[n*64+k+1]) with
  // k = 4*kc + 2*(lane>>4), n = 16*nt + (lane&15). 8-byte aligned global loads.
  for (int s = threadIdx.x; s < 2048; s += 256) {
    const int nt = s >> 9;
    const int kc = (s >> 5) & 15;
    const int ln = s & 31;
    const int k  = 4 * kc + 2 * (ln >> 4);
    const int n  = 16 * nt + (ln & 15);
    sB[s] = *(const float2*)(W + n * N_DIM + k);
  }
  __syncthreads();

  const int wave = threadIdx.x >> 5;
  const int lane = threadIdx.x & 31;
  const int hi   = lane >> 4;
  const int lo   = lane & 15;
  const int row0 = blockIdx.x * BM + wave * 16;
  float* swap = sSwap + wave * 16 * SWSTRIDE;   // wave-private: no barriers needed

  // C/D-layout element for (nt, r): row_local = r + 8*hi, col = 16*nt + lo
  v8f xc[4], z0c[4], dc[4], gc[4], del[4], acc[4];

  // ---- z0 = tanh(x); stage z0 into A-layout for GEMM #1 ----
#pragma unroll
  for (int nt = 0; nt < 4; ++nt) {
#pragma unroll
    for (int r = 0; r < 8; ++r) {
      const float xv = x[(row0 + r + 8 * hi) * N_DIM + 16 * nt + lo];
      xc[nt][r]  = xv;
      z0c[nt][r] = fast_tanh(xv);
      swap[(r + 8 * hi) * SWSTRIDE + 16 * nt + lo] = z0c[nt][r];
    }
  }

  // ---- u = x + z0 @ W^T ; t = tanh(u) ; d = 1 - t^2 ; g = z0 - t ----
#pragma unroll
  for (int nt = 0; nt < 4; ++nt)
#pragma unroll
    for (int r = 0; r < 8; ++r) acc[nt][r] = 0.0f;
  gemm_16x64(swap, sB, lane, acc);

#pragma unroll
  for (int nt = 0; nt < 4; ++nt) {
#pragma unroll
    for (int r = 0; r < 8; ++r) {
      const float u = xc[nt][r] + acc[nt][r];
      const float t = fast_tanh(u);
      dc[nt][r]  = 1.0f - t * t;
      gc[nt][r]  = z0c[nt][r] - t;
      del[nt][r] = gc[nt][r];
      swap[(r + 8 * hi) * SWSTRIDE + 16 * nt + lo] = gc[nt][r];  // delta_0 = g
    }
  }

  // ---- Richardson/Neumann solve of (I - D W) delta = g :
  //      delta <- g + d .* (delta @ W^T)   (converges: rho(DW) < 1) ----
  for (int it = 0; it < KRICH; ++it) {
    v8f a2[4];
#pragma unroll
    for (int nt = 0; nt < 4; ++nt)
#pragma unroll
      for (int r = 0; r < 8; ++r) a2[nt][r] = 0.0f;
    gemm_16x64(swap, sB, lane, a2);   // reads delta_it (LDS in-order per wave)
#pragma unroll
    for (int nt = 0; nt < 4; ++nt) {
#pragma unroll
      for (int r = 0; r < 8; ++r) {
        del[nt][r] = gc[nt][r] + dc[nt][r] * a2[nt][r];
        swap[(r + 8 * hi) * SWSTRIDE + 16 * nt + lo] = del[nt][r];
      }
    }
  }

  // ---- z1 = z0 - delta ; out = tanh(x + z1 @ W^T) ----
#pragma unroll
  for (int nt = 0; nt < 4; ++nt)
#pragma unroll
    for (int r = 0; r < 8; ++r)
      swap[(r + 8 * hi) * SWSTRIDE + 16 * nt + lo] = z0c[nt][r] - del[nt][r];

  v8f a3[4];
#pragma unroll
  for (int nt = 0; nt < 4; ++nt)
#pragma unroll
    for (int r = 0; r < 8; ++r) a3[nt][r] = 0.0f;
  gemm_16x64(swap, sB, lane, a3);

#pragma unroll
  for (int nt = 0; nt < 4; ++nt) {
#pragma unroll
    for (int r = 0; r < 8; ++r) {
      out[(row0 + r + 8 * hi) * N_DIM + 16 * nt + lo] =
          fast_tanh(xc[nt][r] + a3[nt][r]);
    }
  }
}

extern "C" void kernel_launch(void* const* d_in, const int* in_sizes, int n_in,
                              void* d_out, int out_size, void* d_ws, size_t ws_size,
                              hipStream_t stream) {
  (void)n_in; (void)out_size; (void)d_ws; (void)ws_size;
  const float* x = (const float*)d_in[0];
  const float* W = (const float*)d_in[1];
  float* out     = (float*)d_out;
  const int rows = in_sizes[0] / N_DIM;   // 16384
  const int grid = rows / BM;             // 128 workgroups of 256 threads (8 waves)
  hipLaunchKernelGGL(FixedPointLayer_newton_wmma, dim3(grid), dim3(256), 0, stream,
                     x, W, out, rows);
}